// MertForNER_89773406421363
// MI455X (gfx1250) — compile-verified
//
#include <hip/hip_runtime.h>
#include <hip/hip_bf16.h>
#include <math.h>

// Problem constants (from reference): B=64, L=512, H=768, T=9
#define CB 64
#define CL 512
#define CH 768
#define CT 9
#define NPAD 16

typedef __attribute__((ext_vector_type(2))) float v2f;
typedef __attribute__((ext_vector_type(8))) float v8f;

// ---------------------------------------------------------------------------
// Kernel 1: logits = hs @ W^T + b   (32768x768 * 768x9, fp32 WMMA 16x16x4)
// One wave -> 16-row x 16-col tile (cols 9..15 are zero padding).
// 8 waves / block -> 128 rows / block -> 256 blocks.
// ---------------------------------------------------------------------------
__global__ __launch_bounds__(256) void gemm_logits_kernel(
    const float* __restrict__ hs,     // (B*L, H)
    const float* __restrict__ W,      // (T, H)
    const float* __restrict__ bias,   // (T)
    float* __restrict__ out)          // (B*L, T)
{
    // Wp[kk][n] = { W[n][2kk], W[n][2kk+1] }, zero for n >= T.  48 KB LDS.
    __shared__ v2f Wp[CH / 2][NPAD];

    const int tid = threadIdx.x;
    for (int idx = tid; idx < (CH / 2) * NPAD; idx += 256) {
        const int kk = idx >> 4;
        const int n  = idx & 15;
        v2f w; w.x = 0.0f; w.y = 0.0f;
        if (n < CT) {
            w.x = W[n * CH + 2 * kk + 0];
            w.y = W[n * CH + 2 * kk + 1];
        }
        Wp[kk][n] = w;
    }
    __syncthreads();

    const int lane = tid & 31;
    const int wv   = tid >> 5;        // wave in block: 0..7
    const int mrow = lane & 15;       // row within tile (A) / col N (B,D)
    const int hi   = lane >> 4;       // lane-half: selects K pair / row half

    const size_t row0 = ((size_t)blockIdx.x * 8 + (size_t)wv) * 16;
    const float* arow = hs + (row0 + (size_t)mrow) * CH + 2 * hi;

    v8f c = {};   // accumulator, 16x16 fp32

    #pragma unroll 8
    for (int k = 0; k < CH; k += 4) {
        // A fragment: lane(m,hi) holds A[m][k+2*hi], A[m][k+2*hi+1]
        v2f a = *(const v2f*)(arow + k);
        // B fragment: lane(n,hi) holds B[k+2*hi][n], B[k+2*hi+1][n] = W[n][..]
        v2f bf = Wp[(k >> 1) + hi][mrow];
        c = __builtin_amdgcn_wmma_f32_16x16x4_f32(
                /*neg_a=*/false, a, /*neg_b=*/false, bf,
                /*c_mod=*/(short)0, c, /*reuse_a=*/false, /*reuse_b=*/false);
    }

    // D layout: VGPR r -> row (r + 8*hi), col (lane&15)
    const int n = mrow;
    if (n < CT) {
        const float bb = bias[n];
        #pragma unroll
        for (int r = 0; r < 8; ++r) {
            const size_t row = row0 + (size_t)(8 * hi + r);
            out[row * CT + n] = c[r] + bb;
        }
    }
}

// ---------------------------------------------------------------------------
// Kernel 2: per-batch CRF: ws[b] = logZ(b) - score(b)
// One wave (32 threads) per batch; lanes 0..8 own tags.
// ---------------------------------------------------------------------------
__global__ __launch_bounds__(32) void crf_kernel(
    const float* __restrict__ logits,   // (B, L, T)
    const int*   __restrict__ labels,   // (B, L)
    const int*   __restrict__ mask,     // (B, L)
    const float* __restrict__ start_t,  // (T)
    const float* __restrict__ end_t,    // (T)
    const float* __restrict__ trans,    // (T, T)
    float* __restrict__ ws)             // (B) out: logZ - score
{
    const int b    = blockIdx.x;
    const int lane = threadIdx.x;
    const float* lg = logits + (size_t)b * CL * CT;
    const int*   lb = labels + (size_t)b * CL;
    const int*   mk = mask   + (size_t)b * CL;

    // ---- gold path score: parallel over time steps ----
    float part = 0.0f;
    int   mcnt = 0;
    for (int t = lane; t < CL; t += 32) {
        const int lt = lb[t];
        const float em = lg[t * CT + lt];
        if (t == 0) {
            part += start_t[lt] + em;
        } else {
            const float tr = trans[lb[t - 1] * CT + lt];
            part += (em + tr) * (float)mk[t];
        }
        mcnt += (mk[t] != 0) ? 1 : 0;
    }
    for (int o = 16; o > 0; o >>= 1) {
        part += __shfl_xor(part, o, 32);
        mcnt += __shfl_xor(mcnt, o, 32);
    }
    const int seq_end = mcnt - 1;
    const int last    = lb[seq_end];
    const float score = part + end_t[last];

    // ---- forward algorithm (sequential over t) ----
    const int j = (lane < CT) ? lane : 0;
    float trcol[CT];
    #pragma unroll
    for (int i = 0; i < CT; ++i) trcol[i] = trans[i * CT + j];   // column j

    float alpha = start_t[j] + lg[j];   // t = 0

    for (int t = 1; t < CL; ++t) {
        float av[CT];
        #pragma unroll
        for (int i = 0; i < CT; ++i)
            av[i] = __shfl(alpha, i, 32) + trcol[i];
        float m = av[0];
        #pragma unroll
        for (int i = 1; i < CT; ++i) m = fmaxf(m, av[i]);
        float s = 0.0f;
        #pragma unroll
        for (int i = 0; i < CT; ++i) s += __expf(av[i] - m);
        const float nxt = m + __logf(s) + lg[t * CT + j];
        alpha = (mk[t] != 0) ? nxt : alpha;
    }

    // ---- logZ = logsumexp_j(alpha_j + end_t[j]) ----
    float v = (lane < CT) ? (alpha + end_t[j]) : -__builtin_inff();
    float m = v;
    for (int o = 16; o > 0; o >>= 1) m = fmaxf(m, __shfl_xor(m, o, 32));
    float s = (lane < CT) ? __expf(v - m) : 0.0f;
    for (int o = 16; o > 0; o >>= 1) s += __shfl_xor(s, o, 32);
    const float logZ = m + __logf(s);

    if (lane == 0) ws[b] = logZ - score;
}

// ---------------------------------------------------------------------------
// Kernel 3: loss = sum_b ws[b]   (loss = -llh = sum(logZ - score))
// ---------------------------------------------------------------------------
__global__ __launch_bounds__(32) void loss_reduce_kernel(
    const float* __restrict__ ws, float* __restrict__ loss_out)
{
    const int lane = threadIdx.x;
    float v = ws[lane] + ws[lane + 32];
    for (int o = 16; o > 0; o >>= 1) v += __shfl_xor(v, o, 32);
    if (lane == 0) loss_out[0] = v;
}

// ---------------------------------------------------------------------------
extern "C" void kernel_launch(void* const* d_in, const int* in_sizes, int n_in,
                              void* d_out, int out_size, void* d_ws, size_t ws_size,
                              hipStream_t stream) {
    (void)in_sizes; (void)n_in; (void)out_size; (void)ws_size;

    const float* hs      = (const float*)d_in[0];  // (B,L,H)
    const int*   labels  = (const int*)  d_in[1];  // (B,L)
    const int*   amask   = (const int*)  d_in[2];  // (B,L)
    const float* W       = (const float*)d_in[3];  // (T,H)
    const float* bias    = (const float*)d_in[4];  // (T)
    const float* start_t = (const float*)d_in[5];  // (T)
    const float* end_t   = (const float*)d_in[6];  // (T)
    const float* trans   = (const float*)d_in[7];  // (T,T)

    float* logits = (float*)d_out;                           // (B,L,T)
    float* loss   = logits + (size_t)CB * CL * CT;           // scalar
    float* ws     = (float*)d_ws;                            // (B) floats

    // 32768 rows / 128 rows per block = 256 blocks
    gemm_logits_kernel<<<(CB * CL) / 128, 256, 0, stream>>>(hs, W, bias, logits);
    crf_kernel<<<CB, 32, 0, stream>>>(logits, labels, amask,
                                      start_t, end_t, trans, ws);
    loss_reduce_kernel<<<1, 32, 0, stream>>>(ws, loss);
}